// UGP_v3_30030411334316
// MI455X (gfx1250) — compile-verified
//
#include <hip/hip_runtime.h>
#include <math.h>

#define BB 16
#define N_SNPS 500000
#define N_GENES 20000
#define N_NODES 600000
#define N_EDGES 320000
#define NF 8

// 1/sqrt(1 + 1e-5): eval-mode BN with fresh running stats (mean=0, var=1)
#define BN_RSQ 0.99999500003749969f

typedef __attribute__((ext_vector_type(2))) float v2f;
typedef __attribute__((ext_vector_type(8))) float v8f;

// V_WMMA_F32_16X16X4_F32: A 16x4 f32 (2 VGPR), B 4x16 f32 (2 VGPR), C/D 16x16 f32 (8 VGPR)
__device__ __forceinline__ v8f wmma_f32_k4(v2f a, v2f b, v8f c) {
    return __builtin_amdgcn_wmma_f32_16x16x4_f32(false, a, false, b, (short)0, c, false, false);
}

__device__ __forceinline__ float gelu_erf(float x) {
    return 0.5f * x * (1.0f + erff(x * 0.70710678118654752f));
}

// ---------------------------------------------------------------------------
// Kernel 1: SNP -> gene scatter.  h[b][g][f] += snp[b][sid] * filters[f][sid]
// ---------------------------------------------------------------------------
__global__ void node_scatter_kernel(const float* __restrict__ snp,
                                    const int* __restrict__ snp_ids,
                                    const int* __restrict__ gene_of_node,
                                    const float* __restrict__ filters,
                                    float* __restrict__ h) {
    int t = blockIdx.x * blockDim.x + threadIdx.x;
    if (t >= BB * N_NODES) return;
    int b = t / N_NODES;
    int node = t - b * N_NODES;
    int sid = snp_ids[node];
    int g = gene_of_node[node];
    float s = snp[b * N_SNPS + sid];
    float* hp = h + (b * N_GENES + g) * NF;
#pragma unroll
    for (int f = 0; f < NF; ++f)
        atomicAdd(hp + f, s * filters[f * N_SNPS + sid]);
}

// ---------------------------------------------------------------------------
// Kernel 2: edge aggregation.  agg[b][dst][:] += h[b][src][:]
// ---------------------------------------------------------------------------
__global__ void edge_agg_kernel(const float* __restrict__ h,
                                const int* __restrict__ edge_src,
                                const int* __restrict__ edge_dst,
                                float* __restrict__ agg) {
    int t = blockIdx.x * blockDim.x + threadIdx.x;
    if (t >= BB * N_EDGES) return;
    int b = t / N_EDGES;
    int e = t - b * N_EDGES;
    int src = edge_src[e];
    int dst = edge_dst[e];
    const float4* hp = reinterpret_cast<const float4*>(h + (b * N_GENES + src) * NF);
    float4 x0 = hp[0];
    float4 x1 = hp[1];
    float* ap = agg + (b * N_GENES + dst) * NF;
    atomicAdd(ap + 0, x0.x); atomicAdd(ap + 1, x0.y);
    atomicAdd(ap + 2, x0.z); atomicAdd(ap + 3, x0.w);
    atomicAdd(ap + 4, x1.x); atomicAdd(ap + 5, x1.y);
    atomicAdd(ap + 6, x1.z); atomicAdd(ap + 7, x1.w);
}

// ---------------------------------------------------------------------------
// Kernel 3: GIN layer MLP over genes, in-place on h. One wave = one (b, 16-gene tile).
//   z  = (1+eps)*h + agg                         [16 x 8]
//   z1 = gelu(bn(z @ W1 + b1))                   [16 x 16]   (WMMA x2, K=8)
//   h  = gelu(bn(z1 @ W2 + b2))                  [16 x 8]    (WMMA x4, K=16, N padded)
// ---------------------------------------------------------------------------
__global__ void gin_layer_kernel(float* h, const float* __restrict__ agg,
                                 const float* __restrict__ eps, int l,
                                 const float* __restrict__ W1l, const float* __restrict__ b1l,
                                 const float* __restrict__ g1l, const float* __restrict__ bt1l,
                                 const float* __restrict__ W2l, const float* __restrict__ b2l,
                                 const float* __restrict__ g2l, const float* __restrict__ bt2l) {
    __shared__ float stage[4][16 * 17];
    int wid = threadIdx.x >> 5;
    int lane = threadIdx.x & 31;
    int gw = blockIdx.x * 4 + wid;
    int b = gw / (N_GENES / 16);
    int tile = gw - b * (N_GENES / 16);
    if (b >= BB) return;
    int g0 = tile * 16;
    int mn = lane & 15;       // A: row M; B/C: col N
    int khalf = lane >> 4;    // which half of each K=4 chunk this lane holds
    int kb = khalf * 2;

    float e = 1.0f + eps[l];
    int rowoff = (b * N_GENES + g0 + mn) * NF;
    float2 h0 = *(const float2*)(h + rowoff + kb);
    float2 h1 = *(const float2*)(h + rowoff + 4 + kb);
    float2 A0 = *(const float2*)(agg + rowoff + kb);
    float2 A1 = *(const float2*)(agg + rowoff + 4 + kb);
    v2f a0; a0.x = e * h0.x + A0.x; a0.y = e * h0.y + A0.y;
    v2f a1; a1.x = e * h1.x + A1.x; a1.y = e * h1.y + A1.y;

    v2f w10; w10.x = W1l[kb * 16 + mn];       w10.y = W1l[(kb + 1) * 16 + mn];
    v2f w11; w11.x = W1l[(kb + 4) * 16 + mn]; w11.y = W1l[(kb + 5) * 16 + mn];

    v8f c = {};
    c = wmma_f32_k4(a0, w10, c);
    c = wmma_f32_k4(a1, w11, c);

    float sc = g1l[mn] * BN_RSQ, sh = bt1l[mn], bi = b1l[mn];
#pragma unroll
    for (int v = 0; v < 8; ++v) {
        float x = (c[v] + bi) * sc + sh;               // bias + BN
        stage[wid][(v + 8 * khalf) * 17 + mn] = gelu_erf(x);
    }
    __builtin_amdgcn_wave_barrier();

    v8f c2 = {};
#pragma unroll
    for (int kc = 0; kc < 4; ++kc) {
        int kk = kc * 4 + kb;
        v2f a; a.x = stage[wid][mn * 17 + kk]; a.y = stage[wid][mn * 17 + kk + 1];
        v2f w; w.x = (mn < 8) ? W2l[kk * 8 + mn] : 0.0f;
        w.y = (mn < 8) ? W2l[(kk + 1) * 8 + mn] : 0.0f;
        c2 = wmma_f32_k4(a, w, c2);
    }
    if (mn < 8) {
        float sc2 = g2l[mn] * BN_RSQ, sh2 = bt2l[mn], bi2 = b2l[mn];
#pragma unroll
        for (int v = 0; v < 8; ++v) {
            int M = v + 8 * khalf;
            float x = (c2[v] + bi2) * sc2 + sh2;
            h[(b * N_GENES + g0 + M) * NF + mn] = gelu_erf(x);
        }
    }
}

// ---------------------------------------------------------------------------
// Kernel 4: attentive readout.  One WMMA pair computes keys (cols 0-7) and
// values (cols 8-15) simultaneously via packed [Wk | Wv] B operand.
//   w[g]   = sigmoid((h@Wk + bk) @ Wq)
//   g_h[f]+= (h@Wv + bv)[g][f] * w[g]
// ---------------------------------------------------------------------------
__global__ void readout_kernel(const float* __restrict__ h,
                               const float* __restrict__ Wk_, const float* __restrict__ bk_,
                               const float* __restrict__ Wq_,
                               const float* __restrict__ Wv_, const float* __restrict__ bv_,
                               float* __restrict__ w_out, float* __restrict__ g_h) {
    __shared__ float stage[8][16 * 17];
    __shared__ float gacc[NF];
    int wid = threadIdx.x >> 5, lane = threadIdx.x & 31;
    int b = blockIdx.y;
    int mn = lane & 15, khalf = lane >> 4;
    int kb = khalf * 2;

    v2f B0, B1;  // packed [Wk | Wv] B operand (constant across tiles)
    B0.x = (mn < 8) ? Wk_[kb * 8 + mn]       : Wv_[kb * 8 + (mn - 8)];
    B0.y = (mn < 8) ? Wk_[(kb + 1) * 8 + mn] : Wv_[(kb + 1) * 8 + (mn - 8)];
    B1.x = (mn < 8) ? Wk_[(kb + 4) * 8 + mn] : Wv_[(kb + 4) * 8 + (mn - 8)];
    B1.y = (mn < 8) ? Wk_[(kb + 5) * 8 + mn] : Wv_[(kb + 5) * 8 + (mn - 8)];

    float acc[NF];
#pragma unroll
    for (int f = 0; f < NF; ++f) acc[f] = 0.0f;

    int wave_global = blockIdx.x * 8 + wid;
    for (int tile = wave_global; tile < N_GENES / 16; tile += gridDim.x * 8) {
        int g0 = tile * 16;
        int rowoff = (b * N_GENES + g0 + mn) * NF;
        float2 h0 = *(const float2*)(h + rowoff + kb);
        float2 h1 = *(const float2*)(h + rowoff + 4 + kb);
        v2f a0; a0.x = h0.x; a0.y = h0.y;
        v2f a1; a1.x = h1.x; a1.y = h1.y;
        v8f c = {};
        c = wmma_f32_k4(a0, B0, c);
        c = wmma_f32_k4(a1, B1, c);
#pragma unroll
        for (int v = 0; v < 8; ++v)
            stage[wid][(v + 8 * khalf) * 17 + mn] = c[v];
        __builtin_amdgcn_wave_barrier();
        if (lane < 16) {
            float s = 0.0f;
#pragma unroll
            for (int f = 0; f < NF; ++f)
                s += (stage[wid][lane * 17 + f] + bk_[f]) * Wq_[f];
            float wgt = 1.0f / (1.0f + expf(-s));
            w_out[b * N_GENES + g0 + lane] = wgt;
#pragma unroll
            for (int f = 0; f < NF; ++f)
                acc[f] += (stage[wid][lane * 17 + 8 + f] + bv_[f]) * wgt;
        }
        __builtin_amdgcn_wave_barrier();
    }

    if (threadIdx.x < NF) gacc[threadIdx.x] = 0.0f;
    __syncthreads();
    if ((threadIdx.x & 31) < 16) {
#pragma unroll
        for (int f = 0; f < NF; ++f) atomicAdd(&gacc[f], acc[f]);
    }
    __syncthreads();
    if (threadIdx.x < NF) atomicAdd(&g_h[b * NF + threadIdx.x], gacc[threadIdx.x]);
}

// ---------------------------------------------------------------------------
// Kernel 5: final MLP on g_h [16 x 8] -> preds [16 x 1].  One wave, WMMA for stages 1-2.
// ---------------------------------------------------------------------------
__global__ void final_mlp_kernel(const float* __restrict__ g_h,
                                 const float* __restrict__ Wp1, const float* __restrict__ bp1,
                                 const float* __restrict__ gp1, const float* __restrict__ btp1,
                                 const float* __restrict__ Wp2, const float* __restrict__ bp2,
                                 const float* __restrict__ gp2, const float* __restrict__ btp2,
                                 const float* __restrict__ Wp3, const float* __restrict__ bp3,
                                 float* __restrict__ preds) {
    __shared__ float z1s[16][65];
    __shared__ float z2s[16][17];
    int lane = threadIdx.x;
    int mn = lane & 15, khalf = lane >> 4;
    int kb = khalf * 2;

    const float* gp = g_h + mn * NF;
    float2 lo = *(const float2*)(gp + kb);
    float2 hi = *(const float2*)(gp + 4 + kb);
    v2f a0; a0.x = lo.x; a0.y = lo.y;
    v2f a1; a1.x = hi.x; a1.y = hi.y;

    // stage 1: [16x8] @ Wp1[8x64] -> 4 N-tiles
#pragma unroll
    for (int n0 = 0; n0 < 64; n0 += 16) {
        int col = n0 + mn;
        v2f w0; w0.x = Wp1[kb * 64 + col];       w0.y = Wp1[(kb + 1) * 64 + col];
        v2f w1; w1.x = Wp1[(kb + 4) * 64 + col]; w1.y = Wp1[(kb + 5) * 64 + col];
        v8f c = {};
        c = wmma_f32_k4(a0, w0, c);
        c = wmma_f32_k4(a1, w1, c);
        float sc = gp1[col] * BN_RSQ, sh = btp1[col], bi = bp1[col];
#pragma unroll
        for (int v = 0; v < 8; ++v) {
            float x = (c[v] + bi) * sc + sh;
            z1s[v + 8 * khalf][col] = gelu_erf(x);
        }
    }
    __builtin_amdgcn_wave_barrier();

    // stage 2: [16x64] @ Wp2[64x16]
    v8f c2 = {};
#pragma unroll
    for (int kc = 0; kc < 16; ++kc) {
        int k0 = kc * 4 + kb;
        v2f a; a.x = z1s[mn][k0]; a.y = z1s[mn][k0 + 1];
        v2f w; w.x = Wp2[k0 * 16 + mn]; w.y = Wp2[(k0 + 1) * 16 + mn];
        c2 = wmma_f32_k4(a, w, c2);
    }
    {
        float sc = gp2[mn] * BN_RSQ, sh = btp2[mn], bi = bp2[mn];
#pragma unroll
        for (int v = 0; v < 8; ++v) {
            float x = (c2[v] + bi) * sc + sh;
            z2s[v + 8 * khalf][mn] = gelu_erf(x);
        }
    }
    __builtin_amdgcn_wave_barrier();

    // stage 3: [16x16] @ Wp3[16x1]
    if (lane < 16) {
        float s = bp3[0];
#pragma unroll
        for (int n = 0; n < 16; ++n) s += z2s[lane][n] * Wp3[n];
        preds[lane] = s;
    }
}

// ---------------------------------------------------------------------------
extern "C" void kernel_launch(void* const* d_in, const int* in_sizes, int n_in,
                              void* d_out, int out_size, void* d_ws, size_t ws_size,
                              hipStream_t stream) {
    (void)in_sizes; (void)n_in; (void)out_size; (void)ws_size;
    const float* snp          = (const float*)d_in[0];
    const int*   snp_ids      = (const int*)d_in[1];
    const int*   gene_of_node = (const int*)d_in[2];
    const int*   edge_src     = (const int*)d_in[3];
    const int*   edge_dst     = (const int*)d_in[4];
    const float* filters      = (const float*)d_in[5];
    const float* eps          = (const float*)d_in[6];
    const float* W1  = (const float*)d_in[7];
    const float* b1  = (const float*)d_in[8];
    const float* g1  = (const float*)d_in[9];
    const float* bt1 = (const float*)d_in[10];
    const float* W2  = (const float*)d_in[11];
    const float* b2  = (const float*)d_in[12];
    const float* g2  = (const float*)d_in[13];
    const float* bt2 = (const float*)d_in[14];
    const float* Wk  = (const float*)d_in[15];
    const float* bk  = (const float*)d_in[16];
    const float* Wq  = (const float*)d_in[17];
    const float* Wv  = (const float*)d_in[18];
    const float* bv  = (const float*)d_in[19];
    const float* Wp1 = (const float*)d_in[20];
    const float* bp1 = (const float*)d_in[21];
    const float* gp1 = (const float*)d_in[22];
    const float* btp1= (const float*)d_in[23];
    const float* Wp2 = (const float*)d_in[24];
    const float* bp2 = (const float*)d_in[25];
    const float* gp2 = (const float*)d_in[26];
    const float* btp2= (const float*)d_in[27];
    const float* Wp3 = (const float*)d_in[28];
    const float* bp3 = (const float*)d_in[29];

    // Output: preds [16], filters [8*500000], w [16*20000]
    float* out       = (float*)d_out;
    float* preds_out = out;
    float* filt_out  = out + 16;
    float* w_out     = out + 16 + NF * N_SNPS;

    // Workspace: h [B*G*8], agg [B*G*8], g_h [B*8]
    float* h   = (float*)d_ws;
    float* agg = h + (size_t)BB * N_GENES * NF;
    float* g_h = agg + (size_t)BB * N_GENES * NF;

    hipMemcpyAsync(filt_out, filters, sizeof(float) * NF * N_SNPS,
                   hipMemcpyDeviceToDevice, stream);
    hipMemsetAsync(h, 0, sizeof(float) * BB * N_GENES * NF, stream);

    {
        int total = BB * N_NODES;
        node_scatter_kernel<<<(total + 255) / 256, 256, 0, stream>>>(
            snp, snp_ids, gene_of_node, filters, h);
    }

    for (int l = 0; l < 2; ++l) {
        hipMemsetAsync(agg, 0, sizeof(float) * BB * N_GENES * NF, stream);
        int total = BB * N_EDGES;
        edge_agg_kernel<<<(total + 255) / 256, 256, 0, stream>>>(h, edge_src, edge_dst, agg);
        int waves = BB * (N_GENES / 16);
        gin_layer_kernel<<<(waves + 3) / 4, 128, 0, stream>>>(
            h, agg, eps, l,
            W1 + l * NF * 2 * NF, b1 + l * 2 * NF, g1 + l * 2 * NF, bt1 + l * 2 * NF,
            W2 + l * 2 * NF * NF, b2 + l * NF, g2 + l * NF, bt2 + l * NF);
    }

    hipMemsetAsync(g_h, 0, sizeof(float) * BB * NF, stream);
    readout_kernel<<<dim3(32, BB), 256, 0, stream>>>(h, Wk, bk, Wq, Wv, bv, w_out, g_h);

    final_mlp_kernel<<<1, 32, 0, stream>>>(g_h, Wp1, bp1, gp1, btp1,
                                           Wp2, bp2, gp2, btp2, Wp3, bp3, preds_out);
}